// TemporalTGAT_26671746908318
// MI455X (gfx1250) — compile-verified
//
#include <hip/hip_runtime.h>
#include <hip/hip_bf16.h>

typedef __attribute__((ext_vector_type(2))) float v2f;
typedef __attribute__((ext_vector_type(8))) float v8f;

#define HID  64
#define TDIM 32
#define HO   128   // HEADS*OUT
#define OUTC 64

__device__ __forceinline__ v8f wmma4(v2f a, v2f b, v8f c) {
  // D(16x16 f32) = A(16x4 f32) * B(4x16 f32) + C
  return __builtin_amdgcn_wmma_f32_16x16x4_f32(false, a, false, b, (short)0, c,
                                               false, false);
}

__device__ __forceinline__ v2f mk2(float a, float b) {
  v2f r; r.x = a; r.y = b; return r;
}

// ---------------------------------------------------------------- zero scratch
__global__ void k_zero(float* p, int n) {
  int i = blockIdx.x * blockDim.x + threadIdx.x;
  if (i < n) p[i] = 0.0f;
}

// ------------------------------------------------- node gather + 4 projections
// Each wave owns one 16-column block (j) of all 4 weight matrices in registers
// and streams 16-node tiles: q/k/v to workspace, skip projection into d_out.
// The 4 accumulator chains are interleaved per K-step for WMMA latency hiding.
__global__ __launch_bounds__(128) void k_node(
    const int* __restrict__ uu, const int* __restrict__ iu,
    const float* __restrict__ uemb, const float* __restrict__ iemb,
    const float* __restrict__ Wq, const float* __restrict__ bq,
    const float* __restrict__ Wk, const float* __restrict__ bk,
    const float* __restrict__ Wv, const float* __restrict__ bv,
    const float* __restrict__ Wsk, const float* __restrict__ bsk,
    float* __restrict__ qn, float* __restrict__ kn, float* __restrict__ vn,
    float* __restrict__ out, int U, int N)
{
  const int lane  = threadIdx.x & 31;
  const int l     = lane & 15;
  const int half  = lane >> 4;
  const int wg    = blockIdx.x * (blockDim.x >> 5) + (threadIdx.x >> 5);
  const int nw    = gridDim.x * (blockDim.x >> 5);
  const int j     = wg & 7;               // output column block 0..7
  const int col   = j * 16 + l;
  const int tstride = nw >> 3;
  const int ntiles  = (N + 15) >> 4;

  v2f   B[4][16];
  float bias[4];

  auto loadB = [&](const float* W, const float* bb, v2f* Br, float& bi) {
    bi = bb[col];
#pragma unroll
    for (int kk = 0; kk < 16; ++kk) {
      int kr = kk * 4 + half * 2;
      Br[kk] = mk2(W[kr * HO + col], W[(kr + 1) * HO + col]);
    }
  };
  loadB(Wq, bq, B[0], bias[0]);
  loadB(Wk, bk, B[1], bias[1]);
  loadB(Wv, bv, B[2], bias[2]);
  loadB(Wsk, bsk, B[3], bias[3]);

  for (int tile = wg >> 3; tile < ntiles; tile += tstride) {
    const int row0 = tile << 4;
    int r = row0 + l; if (r > N - 1) r = N - 1;   // clamp, keep EXEC full
    const float* xr = (r < U) ? (uemb + (size_t)uu[r] * HID)
                              : (iemb + (size_t)iu[r - U] * HID);
    v2f A[16];
#pragma unroll
    for (int kk = 0; kk < 16; ++kk)
      A[kk] = *(const v2f*)(xr + kk * 4 + half * 2);

    // 4 independent accumulator chains, interleaved per K-step
    v8f acc[4];
#pragma unroll
    for (int m = 0; m < 4; ++m)
#pragma unroll
      for (int rr = 0; rr < 8; ++rr) acc[m][rr] = bias[m];
#pragma unroll
    for (int kk = 0; kk < 16; ++kk) {
#pragma unroll
      for (int m = 0; m < 4; ++m) acc[m] = wmma4(A[kk], B[m][kk], acc[m]);
    }

    float* dsts[4] = {qn, kn, vn, out};
    const bool full = (row0 + 16 <= N);          // wave-uniform
    if (full) {
#pragma unroll
      for (int m = 0; m < 4; ++m) {
        float* bp = dsts[m] + (size_t)(row0 + half * 8) * HO + col;
#pragma unroll
        for (int rr = 0; rr < 8; ++rr) bp[(size_t)rr * HO] = acc[m][rr];
      }
    } else {
#pragma unroll
      for (int m = 0; m < 4; ++m) {
#pragma unroll
        for (int rr = 0; rr < 8; ++rr) {
          int rowm = row0 + rr + half * 8;
          if (rowm < N) dsts[m][(size_t)rowm * HO + col] = acc[m][rr];
        }
      }
    }
  }
}

// ---------------------------------------------- shared edge-tile computation:
// sin(t*scale) -> GEMM1(32x32,+bias) -> LDS transpose -> GEMM2(32x128) -> LDS
// Weights held in registers; each wave has a private LDS region (no barriers,
// EXEC stays all-ones through all WMMAs).
struct EdgeB {
  v2f  Bt[2][8];
  float tbias[2];
  v2f  Be[8][8];
  v2f  SC[8];
};

__device__ __forceinline__ void edge_loadB(
    EdgeB& eb, const float* __restrict__ tW, const float* __restrict__ tb,
    const float* __restrict__ We, int l, int half)
{
  const float LN10_D = 2.302585093f / 32.0f;
#pragma unroll
  for (int j = 0; j < 2; ++j) {
    eb.tbias[j] = tb[j * 16 + l];
#pragma unroll
    for (int kk = 0; kk < 8; ++kk) {
      int kr = kk * 4 + half * 2;
      eb.Bt[j][kk] = mk2(tW[kr * TDIM + j * 16 + l],
                         tW[(kr + 1) * TDIM + j * 16 + l]);
    }
  }
#pragma unroll
  for (int j = 0; j < 8; ++j)
#pragma unroll
    for (int kk = 0; kk < 8; ++kk) {
      int kr = kk * 4 + half * 2;
      eb.Be[j][kk] = mk2(We[kr * HO + j * 16 + l],
                         We[(kr + 1) * HO + j * 16 + l]);
    }
#pragma unroll
  for (int kk = 0; kk < 8; ++kk) {
    int k0 = kk * 4 + half * 2;
    eb.SC[kk] = mk2(__expf(-(float)k0 * LN10_D),
                    __expf(-(float)(k0 + 1) * LN10_D));
  }
}

__device__ __forceinline__ void edge_tile(
    const EdgeB& eb, float t, float* eat, float* et, int l, int half)
{
  v2f A[8];
#pragma unroll
  for (int kk = 0; kk < 8; ++kk)
    A[kk] = mk2(sinf(t * eb.SC[kk].x), sinf(t * eb.SC[kk].y));
  // GEMM1: eattr = sin @ time_W + time_b  -> LDS (row-major 16x32)
#pragma unroll
  for (int j = 0; j < 2; ++j) {
    v8f acc;
#pragma unroll
    for (int rr = 0; rr < 8; ++rr) acc[rr] = eb.tbias[j];
#pragma unroll
    for (int kk = 0; kk < 8; ++kk) acc = wmma4(A[kk], eb.Bt[j][kk], acc);
#pragma unroll
    for (int rr = 0; rr < 8; ++rr)
      eat[(rr + half * 8) * TDIM + j * 16 + l] = acc[rr];
  }
  // transpose through LDS into A-layout
  v2f A2[8];
#pragma unroll
  for (int kk = 0; kk < 8; ++kk) {
    int k0 = kk * 4 + half * 2;
    A2[kk] = *(const v2f*)&eat[l * TDIM + k0];
  }
  // GEMM2: e = eattr @ We (no bias) -> LDS (row-major 16x128)
#pragma unroll
  for (int j = 0; j < 8; ++j) {
    v8f acc;
#pragma unroll
    for (int rr = 0; rr < 8; ++rr) acc[rr] = 0.0f;
#pragma unroll
    for (int kk = 0; kk < 8; ++kk) acc = wmma4(A2[kk], eb.Be[j][kk], acc);
#pragma unroll
    for (int rr = 0; rr < 8; ++rr)
      et[(rr + half * 8) * HO + j * 16 + l] = acc[rr];
  }
}

// -------------------------------------- edge pass 1: alpha, exp, denominators
__global__ __launch_bounds__(128) void k_edge1(
    const int* __restrict__ ei, const float* __restrict__ te,
    const float* __restrict__ tW, const float* __restrict__ tb,
    const float* __restrict__ We,
    const float* __restrict__ qn, const float* __restrict__ kn,
    float* __restrict__ expv, float* __restrict__ sden, int E)
{
  __shared__ float sm[4 * (16 * TDIM + 16 * HO)];
  const int wid  = threadIdx.x >> 5;
  float* eat = sm + wid * (16 * TDIM + 16 * HO);
  float* et  = eat + 16 * TDIM;
  const int lane = threadIdx.x & 31;
  const int l = lane & 15, half = lane >> 4;

  EdgeB eb;
  edge_loadB(eb, tW, tb, We, l, half);

  const int ntiles = (E + 15) >> 4;
  const int wg = blockIdx.x * (blockDim.x >> 5) + wid;
  const int nw = gridDim.x * (blockDim.x >> 5);

  for (int tile = wg; tile < ntiles; tile += nw) {
    const int e0 = tile << 4;
    int er = e0 + l; if (er > E - 1) er = E - 1;
    edge_tile(eb, te[er], eat, et, l, half);

    // lane = (edge l, head half)
    const int  eg  = e0 + l;
    const bool ok  = eg < E;
    const int  egc = ok ? eg : (E - 1);
    const int  src = ei[egc];
    const int  dst = ei[E + egc];
    const float* qp = qn + (size_t)dst * HO + half * OUTC;
    const float* kp = kn + (size_t)src * HO + half * OUTC;
    const float* ep = et + l * HO + half * OUTC;
    float a = 0.0f;
#pragma unroll
    for (int c = 0; c < OUTC; ++c) a += qp[c] * (kp[c] + ep[c]);
    float ev = __expf(a * 0.125f);          // 1/sqrt(64)
    if (ok) {
      expv[(size_t)eg * 2 + half] = ev;
      atomicAdd(&sden[(size_t)dst * 2 + half], ev);
    }
  }
}

// ------------------------------------ edge pass 2: weighted scatter-aggregate
__global__ __launch_bounds__(128) void k_edge2(
    const int* __restrict__ ei, const float* __restrict__ te,
    const float* __restrict__ tW, const float* __restrict__ tb,
    const float* __restrict__ We,
    const float* __restrict__ vn, const float* __restrict__ expv,
    const float* __restrict__ sden, float* __restrict__ out, int E)
{
  __shared__ float sm[4 * (16 * TDIM + 16 * HO)];
  const int wid  = threadIdx.x >> 5;
  float* eat = sm + wid * (16 * TDIM + 16 * HO);
  float* et  = eat + 16 * TDIM;
  const int lane = threadIdx.x & 31;
  const int l = lane & 15, half = lane >> 4;

  EdgeB eb;
  edge_loadB(eb, tW, tb, We, l, half);

  const int ntiles = (E + 15) >> 4;
  const int wg = blockIdx.x * (blockDim.x >> 5) + wid;
  const int nw = gridDim.x * (blockDim.x >> 5);

  for (int tile = wg; tile < ntiles; tile += nw) {
    const int e0 = tile << 4;
    int er = e0 + l; if (er > E - 1) er = E - 1;
    edge_tile(eb, te[er], eat, et, l, half);

    const int  eg  = e0 + l;
    const bool ok  = eg < E;
    const int  egc = ok ? eg : (E - 1);
    const int  src = ei[egc];
    const int  dst = ei[E + egc];
    const float w = expv[(size_t)egc * 2 + half] /
                    (sden[(size_t)dst * 2 + half] + 1e-16f);
    const float* vp = vn + (size_t)src * HO + half * OUTC;
    const float* ep = et + l * HO + half * OUTC;
    float* op = out + (size_t)dst * HO + half * OUTC;
    if (ok) {
#pragma unroll
      for (int c = 0; c < OUTC; ++c)
        atomicAdd(&op[c], w * (vp[c] + ep[c]));
    }
  }
}

// ---------------------------------------------------------------------- launch
extern "C" void kernel_launch(void* const* d_in, const int* in_sizes, int n_in,
                              void* d_out, int out_size, void* d_ws,
                              size_t ws_size, hipStream_t stream)
{
  const int*   uu   = (const int*)d_in[0];
  const int*   iu   = (const int*)d_in[1];
  const int*   ei   = (const int*)d_in[2];
  const float* te   = (const float*)d_in[3];
  const float* uemb = (const float*)d_in[4];
  const float* iemb = (const float*)d_in[5];
  const float* tW   = (const float*)d_in[6];
  const float* tb   = (const float*)d_in[7];
  const float* Wq   = (const float*)d_in[8];
  const float* bq   = (const float*)d_in[9];
  const float* Wk   = (const float*)d_in[10];
  const float* bk   = (const float*)d_in[11];
  const float* Wv   = (const float*)d_in[12];
  const float* bv   = (const float*)d_in[13];
  const float* We   = (const float*)d_in[14];
  const float* Wsk  = (const float*)d_in[15];
  const float* bsk  = (const float*)d_in[16];

  const int U = in_sizes[0];
  const int I = in_sizes[1];
  const int N = U + I;
  const int E = in_sizes[2] / 2;

  float* ws   = (float*)d_ws;
  float* qn   = ws;
  float* kn   = qn + (size_t)N * HO;
  float* vn   = kn + (size_t)N * HO;
  float* expv = vn + (size_t)N * HO;
  float* sden = expv + (size_t)E * 2;
  float* out  = (float*)d_out;

  const int zn = N * 2;
  k_zero<<<(zn + 255) / 256, 256, 0, stream>>>(sden, zn);

  k_node<<<2048, 128, 0, stream>>>(uu, iu, uemb, iemb, Wq, bq, Wk, bk, Wv, bv,
                                   Wsk, bsk, qn, kn, vn, out, U, N);
  k_edge1<<<1024, 128, 0, stream>>>(ei, te, tW, tb, We, qn, kn, expv, sden, E);
  k_edge2<<<1024, 128, 0, stream>>>(ei, te, tW, tb, We, vn, expv, sden, out, E);
}